// QWTForward_24129126269075
// MI455X (gfx1250) — compile-verified
//
#include <hip/hip_runtime.h>

// ---------------------------------------------------------------------------
// QWT forward on MI455X (gfx1250, wave32).
// Reference degenerates to: D = bicubic_down2(image); 16 scaled copies of D.
// Memory bound: ~252 MB traffic -> ~11 us floor @ 23.3 TB/s.
// Horizontal stencil pass done as V_WMMA_F32_16X16X4_F32 (exact fp32 weights).
// ---------------------------------------------------------------------------

typedef __attribute__((ext_vector_type(2))) float v2f;
typedef __attribute__((ext_vector_type(4))) float v4f;
typedef __attribute__((ext_vector_type(8))) float v8f;

#define STW0 (-0.09375f)   // taps for src offset 0.5, a=-0.75 bicubic
#define STW1 (0.59375f)

// banded stencil weight: d in {0,3} -> STW0, {1,2} -> STW1, else 0
__device__ __forceinline__ float stencil_wt(int d) {
  float w = 0.0f;
  if ((unsigned)d < 4u) w = (d == 1 || d == 2) ? STW1 : STW0;
  return w;
}

#define WPB      8            // waves (= 16x16 output tiles) per block
#define WIN_LDS  (34 * 36)    // halo window: 34 rows x 34 cols (+2 pad cols)
#define T_LDS    (16 * 36)    // vertically filtered intermediate

#define OSZ   12582912        // 16*12*256*256  : one output tensor
#define BSZ   196608          // 3*256*256      : one channel-block inside an output

__global__ __launch_bounds__(256)
void qwt_fused(const float* __restrict__ img,
               const float* __restrict__ gl, const float* __restrict__ gh,
               const float* __restrict__ fl, const float* __restrict__ fh,
               float* __restrict__ out) {
  __shared__ float lds[WPB][WIN_LDS + T_LDS];

  const int lane = threadIdx.x & 31;
  const int wv   = threadIdx.x >> 5;
  const int tile = blockIdx.x * WPB + wv;          // 0..12287
  const int tw   = tile & 15;
  const int th   = (tile >> 4) & 15;
  const int p    = tile >> 8;                      // plane 0..47
  const int n    = p / 3;
  const int ch   = p - 3 * n;

  float* win = &lds[wv][0];
  float* T   = &lds[wv][WIN_LDS];
  const float* plane = img + (size_t)p * (512 * 512);

  // filter sums (uniform; the four 30-tap arrays are L2-resident)
  float sgl = 0.f, sgh = 0.f, sfl = 0.f, sfh = 0.f;
  for (int i = 0; i < 30; ++i) {
    sgl += gl[i]; sgh += gh[i]; sfl += fl[i]; sfh += fh[i];
  }

  // ---- stage clamped 34x34 halo window into LDS --------------------------
  const int r0 = 32 * th - 1;
  const int c0 = 32 * tw - 1;
  for (int idx = lane; idx < 34 * 34; idx += 32) {
    int r  = idx / 34;
    int j  = idx - r * 34;
    int gr = min(max(r0 + r, 0), 511);
    int gc = min(max(c0 + j, 0), 511);
    win[r * 36 + j] = plane[gr * 512 + gc];
  }
  // zero the 2 pad columns (keeps NaN out of the WMMA B operand)
  for (int idx = lane; idx < 34 * 2; idx += 32) {
    int r = idx >> 1;
    win[r * 36 + 34 + (idx & 1)] = 0.0f;
  }

  // ---- vertical 4-tap stride-2 pass (VALU): T[16][36] --------------------
  for (int idx = lane; idx < 16 * 36; idx += 32) {
    int r = idx / 36;
    int j = idx - r * 36;
    const float* cp = &win[(2 * r) * 36 + j];
    T[r * 36 + j] = STW0 * (cp[0] + cp[3 * 36]) + STW1 * (cp[36] + cp[2 * 36]);
  }

  // ---- horizontal pass as WMMA: D^T = Ht^T (A, 16x36) x T^T (B, 36x16) ---
  // A layout (f32 16x4, ISA 7.12.2): v0 = K 0|2, v1 = K 1|3, lanes split M.
  // B mirrors the K striping; B[kk,n] = T[n][4k+kk] -> contiguous pair in LDS.
  const int m    = lane & 15;            // A: output col c ; B: T row
  const int hi   = (lane >> 4) & 1;
  const int koff = hi << 1;              // 0 or 2
  v8f acc = {0.f, 0.f, 0.f, 0.f, 0.f, 0.f, 0.f, 0.f};
#pragma unroll
  for (int k = 0; k < 9; ++k) {
    const int j0 = 4 * k + koff;
    v2f A, B;
    A.x = stencil_wt(j0 - 2 * m);
    A.y = stencil_wt(j0 + 1 - 2 * m);
    B.x = T[m * 36 + j0];
    B.y = T[m * 36 + j0 + 1];
    acc = __builtin_amdgcn_wmma_f32_16x16x4_f32(
        /*neg_a=*/false, A, /*neg_b=*/false, B,
        /*c_mod=*/(short)0, acc, /*reuse_a=*/false, /*reuse_b=*/false);
  }

  // ---- 16 scaled copies, two b128 stores each ----------------------------
  // C layout of D^T: lane holds output row (lane&15), cols 8*hi .. 8*hi+7.
  const int R  = 16 * th + m;            // global output row
  const int Wg = 16 * tw + (hi << 3);    // global output col base
  const size_t rowaddr = ((size_t)((n * 12 + ch) * 256 + R)) * 256 + Wg;

#pragma unroll
  for (int o = 0; o < 4; ++o) {
    const float first_e  = (o < 2) ? sgl : sgh;   // b even
    const float first_o  = (o < 2) ? sfl : sfh;   // b odd
    const float second_l = ((o & 1) == 0) ? sgl : sgh;  // b < 2
    const float second_h = ((o & 1) == 0) ? sfl : sfh;  // b >= 2
#pragma unroll
    for (int b = 0; b < 4; ++b) {
      const float s = ((b & 1) ? first_o : first_e) * ((b < 2) ? second_l : second_h);
      float* dst = out + (size_t)o * OSZ + (size_t)b * BSZ + rowaddr;
      v4f lo = {acc[0] * s, acc[1] * s, acc[2] * s, acc[3] * s};
      v4f hv = {acc[4] * s, acc[5] * s, acc[6] * s, acc[7] * s};
      *(v4f*)(dst)     = lo;
      *(v4f*)(dst + 4) = hv;
    }
  }
}

extern "C" void kernel_launch(void* const* d_in, const int* in_sizes, int n_in,
                              void* d_out, int out_size, void* d_ws, size_t ws_size,
                              hipStream_t stream) {
  (void)in_sizes; (void)n_in; (void)out_size; (void)d_ws; (void)ws_size;
  const float* img = (const float*)d_in[0];
  const float* gl  = (const float*)d_in[1];
  const float* gh  = (const float*)d_in[2];
  const float* fl  = (const float*)d_in[3];
  const float* fh  = (const float*)d_in[4];
  float* out = (float*)d_out;

  // 48 planes * 16 * 16 tiles = 12288 tiles, 8 tiles (waves) per 256-thread block
  const int blocks = (48 * 16 * 16) / WPB;   // 1536
  qwt_fused<<<blocks, 256, 0, stream>>>(img, gl, gh, fl, fh, out);
}